// CoPE2d_v2_1769526526341
// MI455X (gfx1250) — compile-verified
//
#include <hip/hip_runtime.h>
#include <hip/hip_bf16.h>

typedef float v2f __attribute__((ext_vector_type(2)));
typedef float v8f __attribute__((ext_vector_type(8)));

#define NG      96          // B*heads
#define SPIX    576         // H*W
#define FEAT    24
#define OUTD    577

// ---------------------------------------------------------------------------
// Stage 1: CoPE feature construction.
// One 256-thread block per unit. blocks [0,2304) = x-direction (unit = (b,h,y)),
// blocks [2304,4608) = y-direction (unit = (b,h,x)).
// Produces X[g][pixel][feat] (fp32) and per-row squared norms.
// ---------------------------------------------------------------------------
__global__ __launch_bounds__(256) void cope_features_kernel(
    const float* __restrict__ q, const float* __restrict__ k,
    const float* __restrict__ pex, const float* __restrict__ pey,
    const int* __restrict__ is_cope_k_p,
    float* __restrict__ Xx, float* __restrict__ Xy,
    float* __restrict__ sqx, float* __restrict__ sqy)
{
    __shared__ float qs[24][64];
    __shared__ float ks[24][64];
    __shared__ float gp[24][24];   // gates, then clamped reverse-cumsum (pos)
    __shared__ float li[24][16];   // logits_int
    __shared__ float xo[24][24];   // unit output

    const int bid = blockIdx.x;
    const int dir = (bid >= 2304) ? 1 : 0;
    const int u   = dir ? (bid - 2304) : bid;
    const int b   = u / 288;          // 12*24
    const int rem = u % 288;
    const int h   = rem / 24;
    const int s   = rem % 24;         // fixed y (dir 0) or fixed x (dir 1)
    const int t   = threadIdx.x;
    const int icopek = *is_cope_k_p;
    const int g = b * 12 + h;

    const float* qb = q + (size_t)g * SPIX * 64;
    const float* kb = k + (size_t)g * SPIX * 64;

    // Stage q/k rows for this unit into LDS
    for (int e = t; e < 24 * 64; e += 256) {
        const int row = e >> 6, d = e & 63;
        const int sidx = (dir == 0) ? (s * 24 + row) : (row * 24 + s);
        qs[row][d] = qb[(size_t)sidx * 64 + d];
        ks[row][d] = kb[(size_t)sidx * 64 + d];
    }
    __syncthreads();

    // gates = sigmoid(SCALE * q_i . k_j)
    for (int e = t; e < 576; e += 256) {
        const int i = e / 24, j = e % 24;
        float acc = 0.f;
        #pragma unroll 8
        for (int d = 0; d < 64; ++d) acc += qs[i][d] * ks[j][d];
        acc *= 0.125f;
        gp[i][j] = 1.0f / (1.0f + __expf(-acc));
    }
    __syncthreads();

    // reverse cumulative sum along j, clamp to NPOS_MAX-1
    if (t < 24) {
        float run = 0.f;
        for (int j = 23; j >= 0; --j) {
            run += gp[t][j];
            gp[t][j] = fminf(run, 15.0f);
        }
    }
    // logits_int[i][p] = src_i . pos_emb[:,p]   (independent of gp; no sync needed yet)
    const float* pe = (dir == 0) ? pex : pey;
    for (int e = t; e < 384; e += 256) {
        const int i = e >> 4, p = e & 15;
        const float* src = icopek ? &qs[i][0] : &ks[i][0];
        float acc = 0.f;
        #pragma unroll 8
        for (int d = 0; d < 64; ++d) acc += src[d] * pe[d * 16 + p];
        li[i][p] = acc;
    }
    __syncthreads();

    // interpolate
    for (int e = t; e < 576; e += 256) {
        const int i = e / 24, j = e % 24;
        const float pos = gp[i][j];
        const float pf  = floorf(pos);
        const int ipf = (int)pf;
        const int ipc = (int)ceilf(pos);
        const float w = pos - pf;
        xo[i][j] = li[i][ipc] * w + li[i][ipf] * (1.0f - w);
    }
    __syncthreads();

    float* X  = (dir == 0) ? Xx  : Xy;
    float* SQ = (dir == 0) ? sqx : sqy;
    for (int e = t; e < 576; e += 256) {
        const int i = e / 24, j = e % 24;
        const int pixel = (dir == 0) ? (s * 24 + i) : (i * 24 + s);
        X[((size_t)g * SPIX + pixel) * FEAT + j] = xo[i][j];
    }
    if (t < 24) {
        float acc = 0.f;
        #pragma unroll
        for (int j = 0; j < 24; ++j) acc += xo[t][j] * xo[t][j];
        const int pixel = (dir == 0) ? (s * 24 + t) : (t * 24 + s);
        SQ[(size_t)g * SPIX + pixel] = acc;
    }
}

// ---------------------------------------------------------------------------
// Zero the CLS pad (row 0 + column 0) of each [577,577] slice.
// ---------------------------------------------------------------------------
__global__ __launch_bounds__(256) void zero_pad_kernel(float* __restrict__ out)
{
    const int idx = blockIdx.x * 256 + threadIdx.x;
    const int per = OUTD + (OUTD - 1);        // 1153
    if (idx >= NG * per) return;
    const int g = idx / per, r = idx % per;
    float* base = out + (size_t)g * OUTD * OUTD;
    if (r < OUTD) base[r] = 0.f;                           // row 0
    else          base[(size_t)(r - OUTD + 1) * OUTD] = 0.f; // col 0
}

// ---------------------------------------------------------------------------
// Stage 2: pairwise distances via fp32 WMMA gram matrices.
// grid = NG * 36 blocks, 128 threads (4 waves). Block owns a 16-row strip of
// the 576x576 distance map for one (b,h); each wave covers 9 column tiles.
// gram = A(16x24) x B(24x16) via 6x v_wmma_f32_16x16x4_f32 per direction.
// ---------------------------------------------------------------------------
__global__ __launch_bounds__(128) void cope_cdist_kernel(
    const float* __restrict__ Xx, const float* __restrict__ Xy,
    const float* __restrict__ sqx, const float* __restrict__ sqy,
    float* __restrict__ out)
{
    const int g     = blockIdx.x / 36;
    const int strip = blockIdx.x % 36;
    const int i0    = strip * 16;
    const int lane  = threadIdx.x & 31;
    const int wave  = threadIdx.x >> 5;      // 0..3
    const int ln16  = lane & 15;
    const int hi    = lane >> 4;             // half-wave select
    const int koff  = hi * 2;                // K, K+1 (lo half) / K+2, K+3 (hi half)

    const float* Xxg  = Xx  + (size_t)g * SPIX * FEAT;
    const float* Xyg  = Xy  + (size_t)g * SPIX * FEAT;
    const float* sqxg = sqx + (size_t)g * SPIX;
    const float* sqyg = sqy + (size_t)g * SPIX;

    // A fragments for this strip (16x4 fp32 layout: lane%16 = M, v2f = {K,K+1})
    v2f ax[6], ay[6];
    {
        const float* rx = Xxg + (size_t)(i0 + ln16) * FEAT;
        const float* ry = Xyg + (size_t)(i0 + ln16) * FEAT;
        #pragma unroll
        for (int kc = 0; kc < 6; ++kc) {
            ax[kc] = *(const v2f*)(rx + kc * 4 + koff);
            ay[kc] = *(const v2f*)(ry + kc * 4 + koff);
        }
    }
    // squared norms for this lane's 8 output rows (C layout: M = r + hi*8)
    float sxi[8], syi[8];
    #pragma unroll
    for (int r = 0; r < 8; ++r) {
        const int irow = i0 + r + hi * 8;
        sxi[r] = sqxg[irow];
        syi[r] = sqyg[irow];
    }

    for (int tt = 0; tt < 9; ++tt) {
        const int j0   = (wave * 9 + tt) * 16;
        const int jcol = j0 + ln16;
        const float* rxB = Xxg + (size_t)jcol * FEAT;
        const float* ryB = Xyg + (size_t)jcol * FEAT;

        v8f gx = {0.f, 0.f, 0.f, 0.f, 0.f, 0.f, 0.f, 0.f};
        v8f gy = {0.f, 0.f, 0.f, 0.f, 0.f, 0.f, 0.f, 0.f};
        #pragma unroll
        for (int kc = 0; kc < 6; ++kc) {
            const v2f bx = *(const v2f*)(rxB + kc * 4 + koff);
            gx = __builtin_amdgcn_wmma_f32_16x16x4_f32(
                    false, ax[kc], false, bx, (short)0, gx, false, false);
        }
        #pragma unroll
        for (int kc = 0; kc < 6; ++kc) {
            const v2f by = *(const v2f*)(ryB + kc * 4 + koff);
            gy = __builtin_amdgcn_wmma_f32_16x16x4_f32(
                    false, ay[kc], false, by, (short)0, gy, false, false);
        }

        const float sxj = sqxg[jcol];
        const float syj = sqyg[jcol];
        float* orow = out + (size_t)g * OUTD * OUTD
                          + (size_t)(i0 + hi * 8 + 1) * OUTD + (1 + jcol);
        #pragma unroll
        for (int r = 0; r < 8; ++r) {
            float d2x = sxi[r] + sxj - 2.0f * gx[r];
            d2x = d2x > 0.f ? d2x : 0.f;
            float d2y = syi[r] + syj - 2.0f * gy[r];
            d2y = d2y > 0.f ? d2y : 0.f;
            orow[(size_t)r * OUTD] = __builtin_sqrtf(d2x) + __builtin_sqrtf(d2y);
        }
    }
}

// ---------------------------------------------------------------------------
extern "C" void kernel_launch(void* const* d_in, const int* in_sizes, int n_in,
                              void* d_out, int out_size, void* d_ws, size_t ws_size,
                              hipStream_t stream)
{
    const float* q   = (const float*)d_in[0];
    const float* k   = (const float*)d_in[1];
    // d_in[2] = value: unused by the reference
    const float* pex = (const float*)d_in[3];
    const float* pey = (const float*)d_in[4];
    const int* icopek = (const int*)d_in[5];
    float* out = (float*)d_out;

    float* ws  = (float*)d_ws;
    const size_t nX = (size_t)NG * SPIX * FEAT;   // 1,327,104 floats each
    float* Xx  = ws;
    float* Xy  = ws + nX;
    float* sqx = ws + 2 * nX;
    float* sqy = ws + 2 * nX + (size_t)NG * SPIX;

    cope_features_kernel<<<4608, 256, 0, stream>>>(q, k, pex, pey, icopek,
                                                   Xx, Xy, sqx, sqy);
    zero_pad_kernel<<<(NG * 1153 + 255) / 256, 256, 0, stream>>>(out);
    cope_cdist_kernel<<<NG * 36, 128, 0, stream>>>(Xx, Xy, sqx, sqy, out);
}